// ResidualVQ_24292335026191
// MI455X (gfx1250) — compile-verified
//
#include <hip/hip_runtime.h>

typedef __attribute__((ext_vector_type(2))) float v2f;
typedef __attribute__((ext_vector_type(8))) float v8f;

#define B_ROWS   262144
#define DIM      128
#define NLEV     3
#define KCB      1024

#define WAVES_PER_WG   8
#define ROWS_PER_WAVE  16
#define ROWS_PER_WG    (WAVES_PER_WG * ROWS_PER_WAVE)   // 128
#define CHUNK_COLS     64
#define NCHUNK         (KCB / CHUNK_COLS)               // 16
#define TILES_PER_CHUNK (CHUNK_COLS / 16)               // 4
#define KSTEPS         (DIM / 4)                        // 32 wmma k-steps
#define LDS_STRIDE     132   // 128 + 4 floats pad -> conflict-free strided b64 LDS reads

// Async DMA: global -> LDS, 16B per lane, tracked by ASYNCcnt (no VGPR round-trip).
__device__ __forceinline__ void async_copy_b128(const float* gsrc, float* lds_dst) {
    unsigned lds_off = (unsigned)(unsigned long long)(uintptr_t)lds_dst; // addr[31:0] = LDS offset
    unsigned long long ga = (unsigned long long)(uintptr_t)gsrc;
    asm volatile("global_load_async_to_lds_b128 %0, %1, off"
                 :: "v"(lds_off), "v"(ga) : "memory");
}
__device__ __forceinline__ void wait_async0() {
    asm volatile("s_wait_asynccnt 0x0" ::: "memory");
}

__global__ __launch_bounds__(256, 1)
void rvq_wmma_kernel(const float* __restrict__ z,
                     const float* __restrict__ codebooks,
                     float* __restrict__ zhat_out,
                     int* __restrict__ idx_out)
{
    __shared__ float lds_cb[2][CHUNK_COLS * LDS_STRIDE];  // double-buffered staged chunks (~67.6 KB)
    __shared__ float lds_cnorm[CHUNK_COLS];
    __shared__ int   lds_idx[WAVES_PER_WG * ROWS_PER_WAVE];

    const int tid  = threadIdx.x;
    const int lane = tid & 31;
    const int wave = tid >> 5;
    const int lh   = lane & 15;                 // row within wave tile / col within 16-col tile
    const int koff = (lane < 16) ? 0 : 2;       // K-pair offset for f32 A/B fragment layout

    const int row = blockIdx.x * ROWS_PER_WG + wave * ROWS_PER_WAVE + lh;

    // Residual = z, held as WMMA A-fragments (16x4 f32 layout) in registers.
    v2f afrag[KSTEPS];
    {
        const float* zr = z + (size_t)row * DIM + koff;
        #pragma unroll
        for (int j = 0; j < KSTEPS; ++j)
            afrag[j] = *(const v2f*)(zr + 4 * j);
    }

    for (int level = 0; level < NLEV; ++level) {
        const float* cb = codebooks + (size_t)level * KCB * DIM;

        float minval[8];
        int   minidx[8];
        #pragma unroll
        for (int v = 0; v < 8; ++v) { minval[v] = 3.4e38f; minidx[v] = 0; }

        // Prologue: kick off async staging of chunk 0 into buffer 0.
        #pragma unroll
        for (int i = 0; i < 8; ++i) {
            int q  = i * 256 + tid;          // float4 id 0..2047
            int r  = q >> 5;                 // local column 0..63
            int kk = (q & 31) << 2;          // k offset 0..124
            async_copy_b128(cb + (size_t)r * DIM + kk, &lds_cb[0][r * LDS_STRIDE + kk]);
        }

        for (int chunk = 0; chunk < NCHUNK; ++chunk) {
            const int cur  = chunk & 1;
            const int col0 = chunk * CHUNK_COLS;

            wait_async0();           // our async writes for `chunk` have landed in LDS
            __syncthreads();         // ...for every wave; and everyone is done with buf[1-cur]

            // Overlap: issue DMA for the next chunk into the other buffer while we compute.
            if (chunk + 1 < NCHUNK) {
                const int coln = col0 + CHUNK_COLS;
                #pragma unroll
                for (int i = 0; i < 8; ++i) {
                    int q  = i * 256 + tid;
                    int r  = q >> 5;
                    int kk = (q & 31) << 2;
                    async_copy_b128(cb + (size_t)(coln + r) * DIM + kk,
                                    &lds_cb[1 - cur][r * LDS_STRIDE + kk]);
                }
            }

            // ||c||^2 per staged column.
            if (tid < CHUNK_COLS) {
                const float* cr = &lds_cb[cur][tid * LDS_STRIDE];
                float s = 0.f;
                #pragma unroll
                for (int k = 0; k < DIM; k += 4) {
                    float4 t = *(const float4*)(cr + k);
                    s += t.x * t.x + t.y * t.y + t.z * t.z + t.w * t.w;
                }
                lds_cnorm[tid] = s;
            }
            __syncthreads();

            #pragma unroll
            for (int tile = 0; tile < TILES_PER_CHUNK; ++tile) {
                // B-fragment source: lane's column row in LDS, K-pair selected by half.
                const float* bptr = &lds_cb[cur][(tile * 16 + lh) * LDS_STRIDE + koff];
                v8f acc = {};
                #pragma unroll
                for (int ks = 0; ks < KSTEPS; ++ks) {
                    v2f b = *(const v2f*)(bptr + 4 * ks);
                    acc = __builtin_amdgcn_wmma_f32_16x16x4_f32(
                        false, afrag[ks], false, b, (short)0, acc, false, false);
                }
                const float cn  = lds_cnorm[tile * 16 + lh];
                const int   col = col0 + tile * 16 + lh;
                #pragma unroll
                for (int v = 0; v < 8; ++v) {
                    float score = cn - 2.0f * acc[v];   // ||r||^2 constant per row: drop it
                    if (score < minval[v]) { minval[v] = score; minidx[v] = col; }
                }
            }
        }

        // Per-row argmin across the 16 lanes of each half (wave32 xor tree),
        // tie-break to smallest index (matches jnp.argmin first-min).
        #pragma unroll
        for (int off = 1; off < 16; off <<= 1) {
            #pragma unroll
            for (int v = 0; v < 8; ++v) {
                float ov = __shfl_xor(minval[v], off, 32);
                int   oi = __shfl_xor(minidx[v], off, 32);
                if (ov < minval[v] || (ov == minval[v] && oi < minidx[v])) {
                    minval[v] = ov; minidx[v] = oi;
                }
            }
        }
        if (lh < 8) {
            int r = lh + ((lane < 16) ? 0 : 8);   // lower half holds rows 0..7, upper 8..15
            lds_idx[wave * 16 + r] = minidx[lh];
        }
        __syncthreads();

        const int mi = lds_idx[wave * 16 + lh];
        if (lane < 16)
            idx_out[(size_t)row * NLEV + level] = mi;

        // residual -= codebook[mi]  (update A-fragments in place)
        const float* cr = cb + (size_t)mi * DIM + koff;
        #pragma unroll
        for (int j = 0; j < KSTEPS; ++j) {
            v2f c = *(const v2f*)(cr + 4 * j);
            afrag[j].x -= c.x;
            afrag[j].y -= c.y;
        }
    }

    // z_hat = z - residual_final (straight-through sum of selected codewords).
    {
        const float* zr = z + (size_t)row * DIM + koff;
        float*       o  = zhat_out + (size_t)row * DIM + koff;
        #pragma unroll
        for (int j = 0; j < KSTEPS; ++j) {
            v2f zv = *(const v2f*)(zr + 4 * j);
            v2f ov;
            ov.x = zv.x - afrag[j].x;
            ov.y = zv.y - afrag[j].y;
            *(v2f*)(o + 4 * j) = ov;
        }
    }
}

extern "C" void kernel_launch(void* const* d_in, const int* in_sizes, int n_in,
                              void* d_out, int out_size, void* d_ws, size_t ws_size,
                              hipStream_t stream) {
    const float* z  = (const float*)d_in[0];                // (B, 128)
    const float* cb = (const float*)d_in[1];                // (3, 1024, 128)
    float* zhat = (float*)d_out;                            // (B, 128) f32
    int*   idx  = (int*)(zhat + (size_t)B_ROWS * DIM);      // (B, 3) i32, concatenated
    dim3 grid(B_ROWS / ROWS_PER_WG);
    rvq_wmma_kernel<<<grid, 256, 0, stream>>>(z, cb, zhat, idx);
}